// GraSP_76613626626433
// MI455X (gfx1250) — compile-verified
//
#include <hip/hip_runtime.h>
#include <hip/hip_bf16.h>
#include <math.h>

// Problem sizes (fixed by the reference)
#define NN  100000
#define EE  1600000
#define GG  128
#define LL  32
#define HH  64
#define KKC 3
#define DDc 256
#define DRc 128
#define TNE 16

typedef __attribute__((ext_vector_type(2))) float v2f;
typedef __attribute__((ext_vector_type(8))) float v8f;

// ---------------------------------------------------------------------------
// FP32 WMMA helper: D = A(16x4) * B(4x16) + C(16x16), wave32, CDNA5
// ---------------------------------------------------------------------------
__device__ __forceinline__ v8f wmma4(v2f a, v2f b, v8f c) {
  return __builtin_amdgcn_wmma_f32_16x16x4_f32(
      /*neg_a=*/false, a, /*neg_b=*/false, b,
      /*c_mod=*/(short)0, c, /*reuse_a=*/false, /*reuse_b=*/false);
}

// ---------------------------------------------------------------------------
// Generic row-major GEMM: C[M,ND] = A[M,KD] @ B[KD,ND] (+bias) (relu) (*rowscale)
// One wave per block; each wave computes a 16 x (16*NT) strip.
// grid = (M/16, ND/(16*NT)), block = 32.
// ---------------------------------------------------------------------------
template <int KD, int ND, int NT, bool RELU, bool BIAS, bool RSCALE>
__global__ __launch_bounds__(32)
void k_gemm(const float* __restrict__ A, const float* __restrict__ B,
            const float* __restrict__ bias, const float* __restrict__ rscale,
            float* __restrict__ C) {
  const int l   = threadIdx.x;
  const int lo  = l & 15;
  const int hi  = l >> 4;          // 0 or 1
  const int row0 = blockIdx.x * 16;
  const int tn0  = blockIdx.y * NT;
  const float* arow = A + (size_t)(row0 + lo) * KD;

  v8f acc[NT];
#pragma unroll
  for (int t = 0; t < NT; ++t) {
    float bv = BIAS ? bias[(tn0 + t) * 16 + lo] : 0.0f;
#pragma unroll
    for (int r = 0; r < 8; ++r) acc[t][r] = bv;
  }

#pragma unroll 2
  for (int k0 = 0; k0 < KD; k0 += 4) {
    const int ka = k0 + 2 * hi;    // lanes 0-15 hold K=k0,k0+1; lanes 16-31 hold K=k0+2,k0+3
    v2f a;
    a.x = arow[ka];
    a.y = arow[ka + 1];
#pragma unroll
    for (int t = 0; t < NT; ++t) {
      const int col = (tn0 + t) * 16 + lo;
      v2f b;
      b.x = B[(size_t)ka * ND + col];
      b.y = B[(size_t)(ka + 1) * ND + col];
      acc[t] = wmma4(a, b, acc[t]);
    }
  }

#pragma unroll
  for (int t = 0; t < NT; ++t) {
    const int col = (tn0 + t) * 16 + lo;
#pragma unroll
    for (int r = 0; r < 8; ++r) {
      const int row = row0 + r + hi * 8;   // C layout: VGPR r -> M=r (lanes<16) / M=r+8
      float v = acc[t][r];
      if (RELU)   v = fmaxf(v, 0.0f);
      if (RSCALE) v *= rscale[row];
      C[(size_t)row * ND + col] = v;
    }
  }
}

// ---------------------------------------------------------------------------
// Attention GEMM fused with per-graph segment-sum:
// ctx[batch[n], :] += (a @ att_W)[n, :]  — never materializes the [N,256] product.
// grid = (N/16, D/64), block = 32.
// ---------------------------------------------------------------------------
__global__ __launch_bounds__(32)
void k_att_scatter(const float* __restrict__ A, const float* __restrict__ B,
                   const int* __restrict__ batch, float* __restrict__ ctx) {
  const int l  = threadIdx.x;
  const int lo = l & 15;
  const int hi = l >> 4;
  const int row0 = blockIdx.x * 16;
  const int tn0  = blockIdx.y * 4;
  const float* arow = A + (size_t)(row0 + lo) * DDc;

  v8f acc[4];
#pragma unroll
  for (int t = 0; t < 4; ++t)
#pragma unroll
    for (int r = 0; r < 8; ++r) acc[t][r] = 0.0f;

#pragma unroll 2
  for (int k0 = 0; k0 < DDc; k0 += 4) {
    const int ka = k0 + 2 * hi;
    v2f a;
    a.x = arow[ka];
    a.y = arow[ka + 1];
#pragma unroll
    for (int t = 0; t < 4; ++t) {
      const int col = (tn0 + t) * 16 + lo;
      v2f b;
      b.x = B[(size_t)ka * DDc + col];
      b.y = B[(size_t)(ka + 1) * DDc + col];
      acc[t] = wmma4(a, b, acc[t]);
    }
  }

#pragma unroll
  for (int r = 0; r < 8; ++r) {
    const int row = row0 + r + hi * 8;
    const int g = batch[row];
#pragma unroll
    for (int t = 0; t < 4; ++t) {
      const int col = (tn0 + t) * 16 + lo;
      atomicAdd(&ctx[(size_t)g * DDc + col], acc[t][r]);
    }
  }
}

// ---------------------------------------------------------------------------
// Small elementwise / scatter kernels
// ---------------------------------------------------------------------------
__global__ void k_deg(const int* __restrict__ dst, float* __restrict__ deg) {
  int e = blockIdx.x * blockDim.x + threadIdx.x;
  if (e < EE) atomicAdd(&deg[dst[e]], 1.0f);
}

__global__ void k_dinv(float* __restrict__ deg) {
  int n = blockIdx.x * blockDim.x + threadIdx.x;
  if (n < NN) deg[n] = rsqrtf(deg[n] + 1.0f);   // +1 self loop
}

__global__ void k_cnt(const int* __restrict__ batch, float* __restrict__ cnt) {
  int n = blockIdx.x * blockDim.x + threadIdx.x;
  if (n < NN) atomicAdd(&cnt[batch[n]], 1.0f);
}

__global__ void k_rowscale(const int* __restrict__ batch, const float* __restrict__ cnt,
                           float* __restrict__ rs) {
  int n = blockIdx.x * blockDim.x + threadIdx.x;
  if (n < NN) rs[n] = rsqrtf(fmaxf(cnt[batch[n]], 1.0f));
}

// a[:, 0:H] = f * (1 + delta)
__global__ void k_scale0(const float* __restrict__ f, const float* __restrict__ delta,
                         float* __restrict__ a) {
  int i = blockIdx.x * blockDim.x + threadIdx.x;
  if (i >= NN * HH) return;
  int n = i / HH, h = i - n * HH;
  a[(size_t)n * DDc + h] = f[i] * (1.0f + delta[0]);
}

// agg = xw * dinv^2 + conv_b   (self-loop term + bias, replaces a memset)
__global__ void k_agg_init(const float* __restrict__ xw, const float* __restrict__ dinv,
                           const float* __restrict__ cb, float* __restrict__ agg) {
  int i = blockIdx.x * blockDim.x + threadIdx.x;
  if (i >= NN * HH) return;
  int n = i / HH, h = i - n * HH;
  float di = dinv[n];
  agg[i] = xw[i] * di * di + cb[h];
}

// agg[dst] += xw[src] * dinv[src]*dinv[dst] : 16 lanes/edge, float4 per lane.
__global__ __launch_bounds__(256)
void k_edge_scatter(const int* __restrict__ src, const int* __restrict__ dst,
                    const float* __restrict__ dinv, const float* __restrict__ xw,
                    float* __restrict__ agg) {
  long long tid = (long long)blockIdx.x * blockDim.x + threadIdx.x;
  long long e = tid >> 4;
  int sub = (int)(tid & 15);
  if (e >= EE) return;
  // stream-ahead prefetch of a future source row (global_prefetch_b8)
  long long ep = e + 4096;
  if (sub == 0 && ep < EE) {
    __builtin_prefetch(xw + (size_t)src[ep] * HH, 0, 1);
  }
  int s = src[e], d = dst[e];
  float coef = dinv[s] * dinv[d];
  const float4* xs = (const float4*)(xw + (size_t)s * HH);
  float4 v = xs[sub];
  float* ad = agg + (size_t)d * HH + sub * 4;
  atomicAdd(ad + 0, v.x * coef);
  atomicAdd(ad + 1, v.y * coef);
  atomicAdd(ad + 2, v.z * coef);
  atomicAdd(ad + 3, v.w * coef);
}

// f = relu(agg) + f ; a[:, colofs:colofs+H] = f
__global__ void k_relu_res_concat(const float* __restrict__ agg, float* __restrict__ f,
                                  float* __restrict__ a, int colofs) {
  int i = blockIdx.x * blockDim.x + threadIdx.x;
  if (i >= NN * HH) return;
  int n = i / HH, h = i - n * HH;
  float v = fmaxf(agg[i], 0.0f) + f[i];
  f[i] = v;
  a[(size_t)n * DDc + colofs + h] = v;
}

// a[:, colofs:colofs+H] = agg (final conv, no relu/residual)
__global__ void k_copy_concat(const float* __restrict__ agg, float* __restrict__ a, int colofs) {
  int i = blockIdx.x * blockDim.x + threadIdx.x;
  if (i >= NN * HH) return;
  int n = i / HH, h = i - n * HH;
  a[(size_t)n * DDc + colofs + h] = agg[i];
}

// ctx = tanh(ctx / max(cnt,1))
__global__ void k_ctx_tanh(float* __restrict__ ctx, const float* __restrict__ cnt) {
  int i = blockIdx.x * blockDim.x + threadIdx.x;
  if (i >= GG * DDc) return;
  int g = i / DDc;
  ctx[i] = tanhf(ctx[i] / fmaxf(cnt[g], 1.0f));
}

// Per node: coef = sigmoid(<a[n], t[batch[n]]>) (wave32 shuffle reduce),
// then p[g] += a[n] * (mu*coef + (1-mu))   — fused mu-blended pooling.
__global__ __launch_bounds__(256)
void k_attcoef_pool(const float* __restrict__ a, const float* __restrict__ t,
                    const int* __restrict__ batch, const float* __restrict__ mu,
                    float* __restrict__ p) {
  int wave = (blockIdx.x * blockDim.x + threadIdx.x) >> 5;
  int lane = threadIdx.x & 31;
  if (wave >= NN) return;
  int g = batch[wave];
  const float* arow = a + (size_t)wave * DDc;
  const float* trow = t + (size_t)g * DDc;
  float s = 0.0f;
  for (int d = lane; d < DDc; d += 32) s += arow[d] * trow[d];
#pragma unroll
  for (int off = 16; off > 0; off >>= 1) s += __shfl_xor(s, off, 32);
  float coef = 1.0f / (1.0f + __expf(-s));
  for (int d = lane; d < DDc; d += 32) {
    float m = mu[d];
    atomicAdd(&p[(size_t)g * DDc + d], arow[d] * (m * coef + (1.0f - m)));
  }
}

// Tensor network: scores[g,k] = relu(gx_g^T W_k hx_g + [gx,hx] V_k + b_k)
__global__ void k_tensor(const float* __restrict__ gx, const float* __restrict__ hx,
                         const float* __restrict__ tnW, const float* __restrict__ tnV,
                         const float* __restrict__ tnb, float* __restrict__ scores) {
  int g = blockIdx.x;
  int k = threadIdx.x;          // 0..15
  const float* gr = gx + (size_t)g * HH;
  const float* hr = hx + (size_t)g * HH;
  float acc = tnb[k];
  for (int e = 0; e < HH; ++e) {
    float wsum = 0.0f;
    for (int d = 0; d < HH; ++d) wsum += gr[d] * tnW[((size_t)d * HH + e) * TNE + k];
    acc += wsum * hr[e];
  }
  float acc2 = 0.0f;
  for (int j = 0; j < HH; ++j) acc2 += tnV[(size_t)k * 2 * HH + j] * gr[j];
  for (int j = 0; j < HH; ++j) acc2 += tnV[(size_t)k * 2 * HH + HH + j] * hr[j];
  scores[g * TNE + k] = fmaxf(acc + acc2, 0.0f);
}

// Final scoring: mlp2 over scores, L2 distance, sigmoid blend.
__global__ void k_final(const float* __restrict__ scores,
                        const float* __restrict__ gx, const float* __restrict__ hx,
                        const float* __restrict__ W1, const float* __restrict__ b1,
                        const float* __restrict__ W2, const float* __restrict__ b2,
                        const float* __restrict__ alpha, float* __restrict__ out) {
  int g = blockIdx.x * blockDim.x + threadIdx.x;
  if (g >= GG) return;
  float h1[TNE];
  for (int k = 0; k < TNE; ++k) {
    float a = b1[k];
    for (int j = 0; j < TNE; ++j) a += scores[g * TNE + j] * W1[j * TNE + k];
    h1[k] = fmaxf(a, 0.0f);
  }
  float sc = b2[0];
  for (int k = 0; k < TNE; ++k) sc += h1[k] * W2[k];
  float l2 = 0.0f;
  for (int d = 0; d < HH; ++d) {
    float df = gx[(size_t)g * HH + d] - hx[(size_t)g * HH + d];
    l2 += df * df;
  }
  l2 = sqrtf(l2);
  float al = alpha[0];
  float s1 = 1.0f / (1.0f + __expf(-sc));
  float s2 = 1.0f / (1.0f + __expf(l2));     // sigmoid(-l2)
  out[g] = al * s1 + (1.0f - al) * s2;
}

// ---------------------------------------------------------------------------
// Orchestration
// ---------------------------------------------------------------------------
extern "C" void kernel_launch(void* const* d_in, const int* in_sizes, int n_in,
                              void* d_out, int out_size, void* d_ws, size_t ws_size,
                              hipStream_t stream) {
  (void)in_sizes; (void)n_in; (void)out_size; (void)ws_size;

  const int*   ei[2]    = {(const int*)d_in[0], (const int*)d_in[1]};
  const int*   batch[2] = {(const int*)d_in[2], (const int*)d_in[3]};
  const float* feat[2]  = {(const float*)d_in[4], (const float*)d_in[5]};
  const float* preW  = (const float*)d_in[6];
  const float* preB  = (const float*)d_in[7];
  const float* convW = (const float*)d_in[8];
  const float* convB = (const float*)d_in[9];
  const float* delta = (const float*)d_in[10];
  const float* alpha = (const float*)d_in[11];
  const float* mu    = (const float*)d_in[12];
  const float* attW  = (const float*)d_in[13];
  const float* pW1   = (const float*)d_in[14];
  const float* pB1   = (const float*)d_in[15];
  const float* pW2   = (const float*)d_in[16];
  const float* pB2   = (const float*)d_in[17];
  const float* tnW   = (const float*)d_in[18];
  const float* tnV   = (const float*)d_in[19];
  const float* tnB   = (const float*)d_in[20];
  const float* scW1  = (const float*)d_in[21];
  const float* scB1  = (const float*)d_in[22];
  const float* scW2  = (const float*)d_in[23];
  const float* scB2  = (const float*)d_in[24];
  float* out = (float*)d_out;

  // Workspace carving (floats). Sides processed sequentially -> big buffers reused.
  float* w = (float*)d_ws;
  size_t o = 0;
  float* dinv = w + o; o += NN;
  float* cnt  = w + o; o += GG;
  float* rs   = w + o; o += NN;
  float* f    = w + o; o += (size_t)NN * HH;
  float* xw   = w + o; o += (size_t)NN * HH;
  float* agg  = w + o; o += (size_t)NN * HH;
  float* abuf = w + o; o += (size_t)NN * DDc;
  float* ctx  = w + o; o += (size_t)GG * DDc;
  float* p1   = w + o; o += (size_t)GG * DDc;
  float* p2   = w + o; o += (size_t)GG * DDc;
  float* gxb  = w + o; o += (size_t)GG * HH;
  float* hxb  = w + o; o += (size_t)GG * HH;
  float* hid  = w + o; o += (size_t)GG * DRc;
  float* scr  = w + o; o += (size_t)GG * TNE;

  float* pside[2] = {p1, p2};
  float* gh[2]    = {gxb, hxb};

  const int TB = 256;
  const int nhBlocks = (NN * HH + TB - 1) / TB;

  for (int s = 0; s < 2; ++s) {
    const int* srcp = ei[s];
    const int* dstp = ei[s] + EE;

    hipMemsetAsync(dinv, 0, NN * sizeof(float), stream);
    hipMemsetAsync(cnt,  0, GG * sizeof(float), stream);
    k_deg<<<(EE + TB - 1) / TB, TB, 0, stream>>>(dstp, dinv);
    k_dinv<<<(NN + TB - 1) / TB, TB, 0, stream>>>(dinv);
    k_cnt<<<(NN + TB - 1) / TB, TB, 0, stream>>>(batch[s], cnt);
    k_rowscale<<<(NN + TB - 1) / TB, TB, 0, stream>>>(batch[s], cnt, rs);

    // f = graph_size_norm(features @ pre_W + pre_b)
    k_gemm<LL, HH, 4, false, true, true>
        <<<dim3(NN / 16, 1), 32, 0, stream>>>(feat[s], preW, preB, rs, f);

    // a[:, 0:H] = f * (1 + delta)
    k_scale0<<<nhBlocks, TB, 0, stream>>>(f, delta, abuf);

    for (int i = 0; i < KKC; ++i) {
      // xw = f @ conv_W[i]
      k_gemm<HH, HH, 4, false, false, false>
          <<<dim3(NN / 16, 1), 32, 0, stream>>>(f, convW + (size_t)i * HH * HH,
                                                nullptr, nullptr, xw);
      // agg = xw*dinv^2 + b  then  agg[dst] += coef * xw[src]
      k_agg_init<<<nhBlocks, TB, 0, stream>>>(xw, dinv, convB + i * HH, agg);
      k_edge_scatter<<<(int)(((long long)EE * 16 + TB - 1) / TB), TB, 0, stream>>>(
          srcp, dstp, dinv, xw, agg);
      if (i < KKC - 1)
        k_relu_res_concat<<<nhBlocks, TB, 0, stream>>>(agg, f, abuf, (i + 1) * HH);
      else
        k_copy_concat<<<nhBlocks, TB, 0, stream>>>(agg, abuf, KKC * HH);
    }

    // Attention pooling
    hipMemsetAsync(ctx, 0, (size_t)GG * DDc * sizeof(float), stream);
    k_att_scatter<<<dim3(NN / 16, DDc / 64), 32, 0, stream>>>(abuf, attW, batch[s], ctx);
    k_ctx_tanh<<<(GG * DDc + TB - 1) / TB, TB, 0, stream>>>(ctx, cnt);
    hipMemsetAsync(pside[s], 0, (size_t)GG * DDc * sizeof(float), stream);
    k_attcoef_pool<<<(NN * 32 + TB - 1) / TB, TB, 0, stream>>>(abuf, ctx, batch[s],
                                                               mu, pside[s]);

    // gx/hx = mlp2(p)
    k_gemm<DDc, DRc, 4, true, true, false>
        <<<dim3(GG / 16, DRc / 64), 32, 0, stream>>>(pside[s], pW1, pB1, nullptr, hid);
    k_gemm<DRc, HH, 4, false, true, false>
        <<<dim3(GG / 16, 1), 32, 0, stream>>>(hid, pW2, pB2, nullptr, gh[s]);
  }

  k_tensor<<<GG, TNE, 0, stream>>>(gh[0], gh[1], tnW, tnV, tnB, scr);
  k_final<<<1, GG, 0, stream>>>(scr, gh[0], gh[1], scW1, scB1, scW2, scB2, alpha, out);
}